// OhemPairTripletLoss_49546742727252
// MI455X (gfx1250) — compile-verified
//
#include <hip/hip_runtime.h>

typedef __attribute__((ext_vector_type(2))) float v2f;
typedef __attribute__((ext_vector_type(8))) float v8f;

#define NROWS 4096
#define DDIM  512
#define MARGIN_F 0.3f
#define KSEL 2048u

struct SelState {
    double   allSum;   // sum of all off-diagonal losses (both matrices)
    double   gtSum;    // sum of losses strictly greater than threshold T
    unsigned prefix;   // current radix-select bit prefix
    unsigned kRem;     // remaining k at current level
    unsigned Tpat;     // final threshold bit pattern (== 2048th largest value)
    unsigned needEq;   // copies of T needed to complete top-k
};

// ---------------------------------------------------------------- init / zero
__global__ void k_init(SelState* st, unsigned* hist) {
    int g = blockIdx.x * blockDim.x + threadIdx.x;
    if (g == 0) {
        st->allSum = 0.0; st->gtSum = 0.0;
        st->prefix = 0u;  st->kRem = KSEL;
        st->Tpat = 0u;    st->needEq = 0u;
    }
    if (g < 4096) hist[g] = 0u;
}

__global__ void k_zhist(unsigned* hist) {
    int g = blockIdx.x * blockDim.x + threadIdx.x;
    if (g < 4096) hist[g] = 0u;
}

// ---------------------------------------------------------------- row norms
// xx[i] = ||p1_i||^2, yy[i] = ||p2_i||^2, ap[i] = ||p1_i - p2_i||^2
__global__ __launch_bounds__(256) void k_norms(const float* __restrict__ p1,
                                               const float* __restrict__ p2,
                                               float* xx, float* yy, float* ap) {
    const int row = blockIdx.x;
    const float* a = p1 + (size_t)row * DDIM;
    const float* b = p2 + (size_t)row * DDIM;
    float sx = 0.f, sy = 0.f, sd = 0.f;
    for (int c = threadIdx.x; c < DDIM; c += 256) {
        float x = a[c], y = b[c];
        sx += x * x;
        sy += y * y;
        float d = x - y;
        sd += d * d;
    }
    __shared__ float rx[256], ry[256], rd[256];
    rx[threadIdx.x] = sx; ry[threadIdx.x] = sy; rd[threadIdx.x] = sd;
    __syncthreads();
    for (int s = 128; s > 0; s >>= 1) {
        if (threadIdx.x < s) {
            rx[threadIdx.x] += rx[threadIdx.x + s];
            ry[threadIdx.x] += ry[threadIdx.x + s];
            rd[threadIdx.x] += rd[threadIdx.x + s];
        }
        __syncthreads();
    }
    if (threadIdx.x == 0) { xx[row] = rx[0]; yy[row] = ry[0]; ap[row] = rd[0]; }
}

// ---------------------------------------------------------------- main pass
// Computes a 128x128 tile of S = A * B^T with V_WMMA_F32_16X16X4_F32, then
// loss[i][j] = relu(ap[i] - relu(xx[i] + yyB[j] - 2 S[i][j]) + margin), i != j.
// mode 0: accumulate allSum + histogram of bits[31:20]
// mode 1: histogram of bits[19:8] where bits[31:20] == prefix
// mode 2: histogram of bits[7:0]  where bits[31:8]  == prefix
// mode 3: accumulate gtSum over values with bit pattern > Tpat
//
// LDS staging stores pre-paired K values (v2f {A[2k][m], A[2k+1][m]}) so each
// WMMA A/B fragment is a single aligned ds_load_b64 into an even VGPR pair.
// Row stride 144 float2 = 288 dwords; 288 % 64 == 32, so the lane 0-15 (row
// kp) and lane 16-31 (row kp+1) halves hit disjoint bank sets.
__global__ __launch_bounds__(256) void k_pass(const float* __restrict__ A,
                                              const float* __restrict__ B,
                                              const float* __restrict__ xx,
                                              const float* __restrict__ yyB,
                                              const float* __restrict__ ap,
                                              SelState* st, unsigned* ghist,
                                              int mode) {
    __shared__ v2f AsP[16][144];      // 16 k-pairs x 128 rows (padded)
    __shared__ v2f BsP[16][144];
    __shared__ unsigned hist[4096];
    __shared__ double red[256];

    const int tid  = threadIdx.x;
    const int lane = tid & 31;
    const int w    = tid >> 5;        // 8 wave32s
    const int wm   = w & 1;           // 2 M-waves -> 64 rows each
    const int wn   = w >> 1;          // 4 N-waves -> 32 cols each
    const int ibase = blockIdx.y * 128;
    const int jbase = blockIdx.x * 128;

    if (mode != 3)
        for (int b = tid; b < 4096; b += 256) hist[b] = 0u;

    unsigned prefixVal = 0u, Tpat = 0u;
    if (mode == 1 || mode == 2) prefixVal = st->prefix;
    if (mode == 3)              Tpat      = st->Tpat;
    __syncthreads();

    v8f acc[4][2];
#pragma unroll
    for (int ti = 0; ti < 4; ++ti)
#pragma unroll
        for (int tj = 0; tj < 2; ++tj)
            acc[ti][tj] = (v8f){0.f, 0.f, 0.f, 0.f, 0.f, 0.f, 0.f, 0.f};

    const int lm  = lane & 15;        // m / n within 16x16 tile
    const int kh  = lane >> 4;        // k-pair half: 0 (K=0,1) or 1 (K=2,3)

    for (int kc = 0; kc < DDIM; kc += 32) {
        // stage 128x32 chunks of A and B rows into LDS as paired k-values
#pragma unroll
        for (int r = 0; r < 4; ++r) {
            int linear = tid + r * 256;          // 0..1023 float4 slots
            int row = linear >> 3;               // 0..127
            int c4  = (linear & 7) << 2;         // k offset 0,4,...,28
            int kp0 = c4 >> 1;                   // k-pair index 0,2,...,14
            const float4 va = *(const float4*)(A + (size_t)(ibase + row) * DDIM + kc + c4);
            AsP[kp0 + 0][row] = (v2f){va.x, va.y};
            AsP[kp0 + 1][row] = (v2f){va.z, va.w};
            const float4 vb = *(const float4*)(B + (size_t)(jbase + row) * DDIM + kc + c4);
            BsP[kp0 + 0][row] = (v2f){vb.x, vb.y};
            BsP[kp0 + 1][row] = (v2f){vb.z, vb.w};
        }
        __syncthreads();

#pragma unroll
        for (int kk = 0; kk < 32; kk += 4) {
            const int kp = (kk >> 1) + kh;
            v2f af[4], bf[2];
#pragma unroll
            for (int ti = 0; ti < 4; ++ti)
                af[ti] = AsP[kp][wm * 64 + ti * 16 + lm];
#pragma unroll
            for (int tj = 0; tj < 2; ++tj)
                bf[tj] = BsP[kp][wn * 32 + tj * 16 + lm];
#pragma unroll
            for (int ti = 0; ti < 4; ++ti)
#pragma unroll
                for (int tj = 0; tj < 2; ++tj)
                    acc[ti][tj] = __builtin_amdgcn_wmma_f32_16x16x4_f32(
                        false, af[ti], false, bf[tj], (short)0, acc[ti][tj],
                        false, false);
        }
        __syncthreads();
    }

    // ----- epilogue: loss + mode-specific accumulation
    double localSum = 0.0;
    const int mtop = kh * 8;          // C layout: rows r / 8+r per lane half
#pragma unroll
    for (int ti = 0; ti < 4; ++ti) {
#pragma unroll
        for (int tj = 0; tj < 2; ++tj) {
            const int j = jbase + wn * 32 + tj * 16 + lm;
            const float yv = yyB[j];
#pragma unroll
            for (int r = 0; r < 8; ++r) {
                const int i = ibase + wm * 64 + ti * 16 + mtop + r;
                if (i == j) continue;
                const float s = acc[ti][tj][r];
                float d = xx[i] + yv - 2.0f * s;
                d = fmaxf(d, 0.0f);
                float loss = ap[i] - d + MARGIN_F;
                loss = fmaxf(loss, 0.0f);
                const unsigned bits = __float_as_uint(loss);
                if (mode == 0) {
                    localSum += (double)loss;
                    atomicAdd(&hist[bits >> 20], 1u);
                } else if (mode == 1) {
                    if ((bits >> 20) == prefixVal)
                        atomicAdd(&hist[(bits >> 8) & 0xFFFu], 1u);
                } else if (mode == 2) {
                    if ((bits >> 8) == prefixVal)
                        atomicAdd(&hist[bits & 0xFFu], 1u);
                } else {
                    if (bits > Tpat) localSum += (double)loss;
                }
            }
        }
    }

    if (mode == 0 || mode == 3) {
        red[tid] = localSum;
        __syncthreads();
        for (int s2 = 128; s2 > 0; s2 >>= 1) {
            if (tid < s2) red[tid] += red[tid + s2];
            __syncthreads();
        }
        if (tid == 0)
            atomicAdd(mode == 0 ? &st->allSum : &st->gtSum, red[0]);
    }
    if (mode != 3) {
        __syncthreads();
        for (int b = tid; b < 4096; b += 256) {
            unsigned c = hist[b];
            if (c) atomicAdd(&ghist[b], c);
        }
    }
}

// ---------------------------------------------------------------- radix scan
__global__ void k_scan(SelState* st, const unsigned* hist, int level) {
    if (threadIdx.x != 0 || blockIdx.x != 0) return;
    const int nb = (level < 2) ? 4096 : 256;
    const unsigned kRem = st->kRem;
    unsigned long long cum = 0;
    int bsel = 0;
    unsigned cAbove = 0;
    for (int b = nb - 1; b >= 0; --b) {
        unsigned long long c = hist[b];
        if (cum + c >= (unsigned long long)kRem) { bsel = b; cAbove = (unsigned)cum; break; }
        cum += c;
    }
    unsigned newPrefix;
    if (level == 0)      newPrefix = (unsigned)bsel;
    else if (level == 1) newPrefix = (st->prefix << 12) | (unsigned)bsel;
    else                 newPrefix = (st->prefix << 8)  | (unsigned)bsel;
    const unsigned newKRem = kRem - cAbove;
    st->prefix = newPrefix;
    st->kRem   = newKRem;
    if (level == 2) { st->Tpat = newPrefix; st->needEq = newKRem; }
}

// ---------------------------------------------------------------- finalize
__global__ void k_final(const SelState* st, float* out) {
    if (threadIdx.x != 0 || blockIdx.x != 0) return;
    const double allLoss =
        st->allSum / (2.0 * (double)NROWS * ((double)NROWS - 1.0));
    const float Tf = __uint_as_float(st->Tpat);
    double sum; unsigned num;
    if (Tf > 0.0f) {
        num = KSEL;
        sum = st->gtSum + (double)st->needEq * (double)Tf;
    } else {
        num = KSEL - st->needEq;   // only strictly-positive entries counted
        sum = st->gtSum;
    }
    const double avg = (num > 0) ? sum / (double)num : 0.0;
    out[0] = (float)avg;      // avg_loss
    out[1] = (float)allLoss;  // all_loss
}

// ---------------------------------------------------------------- launch
extern "C" void kernel_launch(void* const* d_in, const int* in_sizes, int n_in,
                              void* d_out, int out_size, void* d_ws, size_t ws_size,
                              hipStream_t stream) {
    (void)in_sizes; (void)n_in; (void)out_size; (void)ws_size;
    const float* p1 = (const float*)d_in[0];
    const float* p2 = (const float*)d_in[1];
    float* out = (float*)d_out;

    char* ws = (char*)d_ws;
    SelState* st   = (SelState*)ws;                        // 64 B reserved
    unsigned* hist = (unsigned*)(ws + 64);                 // 4096 u32
    float* xx = (float*)(ws + 64 + 4096 * 4);              // 4096 f32
    float* yy = (float*)(ws + 64 + 4096 * 4 + 4096 * 4);
    float* ap = (float*)(ws + 64 + 4096 * 4 + 2 * 4096 * 4);

    const dim3 grid(NROWS / 128, NROWS / 128);

    k_init<<<16, 256, 0, stream>>>(st, hist);
    k_norms<<<NROWS, 256, 0, stream>>>(p1, p2, xx, yy, ap);

    // mode 0: all-loss sum + level-0 histogram (loss1: B=pair1, loss2: B=pair2)
    k_pass<<<grid, 256, 0, stream>>>(p1, p1, xx, xx, ap, st, hist, 0);
    k_pass<<<grid, 256, 0, stream>>>(p1, p2, xx, yy, ap, st, hist, 0);
    k_scan<<<1, 32, 0, stream>>>(st, hist, 0);

    k_zhist<<<16, 256, 0, stream>>>(hist);
    k_pass<<<grid, 256, 0, stream>>>(p1, p1, xx, xx, ap, st, hist, 1);
    k_pass<<<grid, 256, 0, stream>>>(p1, p2, xx, yy, ap, st, hist, 1);
    k_scan<<<1, 32, 0, stream>>>(st, hist, 1);

    k_zhist<<<16, 256, 0, stream>>>(hist);
    k_pass<<<grid, 256, 0, stream>>>(p1, p1, xx, xx, ap, st, hist, 2);
    k_pass<<<grid, 256, 0, stream>>>(p1, p2, xx, yy, ap, st, hist, 2);
    k_scan<<<1, 32, 0, stream>>>(st, hist, 2);

    // mode 3: sum of losses strictly above exact threshold
    k_pass<<<grid, 256, 0, stream>>>(p1, p1, xx, xx, ap, st, hist, 3);
    k_pass<<<grid, 256, 0, stream>>>(p1, p2, xx, yy, ap, st, hist, 3);

    k_final<<<1, 32, 0, stream>>>(st, out);
}